// IntensityTransform_1554778161489
// MI455X (gfx1250) — compile-verified
//
#include <hip/hip_runtime.h>
#include <hip/hip_bf16.h>

// IntensityTransform: out[b,c,h,w] = T[b,c, rint(255*img[b,c,h,w])]
// Bandwidth-bound streaming gather (~201 MB => ~8.6us floor @ 23.3 TB/s).
// CDNA5 paths used: async global->LDS LUT staging (ASYNCcnt), LDS gather,
// non-temporal 128-bit stream loads/stores, wave32-sized blocks.

typedef float v4f __attribute__((ext_vector_type(4)));

#define AS1 __attribute__((address_space(1)))
#define AS3 __attribute__((address_space(3)))

__global__ __launch_bounds__(256) void intensity_lut_kernel(
    const float* __restrict__ images,
    const float* __restrict__ transforms,
    float* __restrict__ out,
    int n4)  // number of float4 elements per (b,c) plane
{
    __shared__ float lut[256];

    const int bc = blockIdx.y;
    const int t  = threadIdx.x;

    const float* lsrc = transforms + (size_t)bc * 256;

    // ---- Stage the 256-entry LUT into LDS via the gfx1250 async path ----
#if __has_builtin(__builtin_amdgcn_global_load_async_to_lds_b32)
    if (t < 256) {
        AS1 int* gp = (AS1 int*)(float*)(lsrc + t);   // addrspacecast generic->global
        AS3 int* lp = (AS3 int*)(&lut[t]);            // addrspacecast generic->LDS
        __builtin_amdgcn_global_load_async_to_lds_b32(gp, lp, /*offset=*/0, /*cpol=*/0);
    }
  #if __has_builtin(__builtin_amdgcn_s_wait_asynccnt)
    __builtin_amdgcn_s_wait_asynccnt(0);
  #else
    asm volatile("s_wait_asynccnt 0" ::: "memory");
  #endif
#else
    if (t < 256) {
        lut[t] = lsrc[t];
    }
#endif
    __syncthreads();

    // ---- Stream the plane: 128-bit non-temporal load -> 4x LDS gather -> 128-bit NT store ----
    const size_t base4 = (size_t)bc * (size_t)n4;
    const v4f* __restrict__ in4  = (const v4f*)images + base4;
    v4f* __restrict__       out4 = (v4f*)out + base4;

    const int stride = gridDim.x * blockDim.x;
    for (int i = blockIdx.x * blockDim.x + t; i < n4; i += stride) {
        v4f v = __builtin_nontemporal_load(in4 + i);

        // jnp.round == round-half-to-even == v_rndne_f32 (default RNE mode)
        int i0 = (int)__builtin_rintf(v.x * 255.0f);
        int i1 = (int)__builtin_rintf(v.y * 255.0f);
        int i2 = (int)__builtin_rintf(v.z * 255.0f);
        int i3 = (int)__builtin_rintf(v.w * 255.0f);

        // Inputs are uniform [0,1) so indices are in range; clamp is 8 cheap VALU
        // ops per 4 pixels, negligible vs the HBM bound, and guards FP edge cases.
        i0 = min(max(i0, 0), 255);
        i1 = min(max(i1, 0), 255);
        i2 = min(max(i2, 0), 255);
        i3 = min(max(i3, 0), 255);

        v4f r;
        r.x = lut[i0];
        r.y = lut[i1];
        r.z = lut[i2];
        r.w = lut[i3];

        __builtin_nontemporal_store(r, out4 + i);
    }
}

extern "C" void kernel_launch(void* const* d_in, const int* in_sizes, int n_in,
                              void* d_out, int out_size, void* d_ws, size_t ws_size,
                              hipStream_t stream) {
    const float* images     = (const float*)d_in[0];  // [B,C,H,W] f32
    const float* transforms = (const float*)d_in[1];  // [B,C,256] f32
    float* out              = (float*)d_out;          // [B,C,H,W] f32

    const int bc = in_sizes[1] / 256;                 // B*C planes (96)
    const long hw = (long)in_sizes[0] / (long)bc;     // H*W per plane (262144)
    const int n4 = (int)(hw / 4);                     // float4s per plane (65536)

    // ~8 float4 per thread => 32 blocks per plane, 3072 blocks total.
    int blocks_x = (n4 + 256 * 8 - 1) / (256 * 8);
    if (blocks_x < 1) blocks_x = 1;

    dim3 grid((unsigned)blocks_x, (unsigned)bc, 1);
    dim3 block(256, 1, 1);
    hipLaunchKernelGGL(intensity_lut_kernel, grid, block, 0, stream,
                       images, transforms, out, n4);
}